// HybridRealQ_21938692948013
// MI455X (gfx1250) — compile-verified
//
#include <hip/hip_runtime.h>

typedef __attribute__((ext_vector_type(2))) float v2f;
typedef __attribute__((ext_vector_type(4))) float v4f;
typedef __attribute__((ext_vector_type(8))) float v8f;

#define WAVES_PER_BLOCK 8
#define TILES_PER_WAVE  8
#define ROWS_PER_TILE   16

// per-wave LDS staging strides (floats); all even (b64/b128 aligned),
// bank-conflict-free mod 64 for the 16-lane column accesses
#define STRIDE_A   20    // 16 x [ x_c(16) | q(1) | pad(3) ]  (K order permuted!)
#define STRIDE_H1  36
#define STRIDE_H2  20
#define A_FLOATS   (16 * STRIDE_A)    // 320
#define H1_FLOATS  (16 * STRIDE_H1)   // 576
#define H2_FLOATS  (16 * STRIDE_H2)   // 320
#define WAVE_LDS   (A_FLOATS + H1_FLOATS + H2_FLOATS)  // 1216 floats

// block-shared weights: W1 stored K-permuted + zero padded to [20][32]:
//   rows 0..15 = W1[1..16] (x_c weights), row 16 = W1[0] (q weight), rows 17..19 = 0
#define SW1_OFF  0
#define SW1_N    (20 * 32)            // 640
#define SW2_OFF  640
#define SW2_N    (32 * 16)            // 512
#define SB1_OFF  1152
#define SB2_OFF  1184
#define SW3_OFF  1200
#define SWT_N    1216

__global__ __launch_bounds__(256)
void hybridq_kernel(const float* __restrict__ x_q,
                    const float* __restrict__ x_c,
                    const float* __restrict__ q_params,
                    const float* __restrict__ W1, const float* __restrict__ b1,
                    const float* __restrict__ W2, const float* __restrict__ b2,
                    const float* __restrict__ W3, const float* __restrict__ b3,
                    float* __restrict__ out)
{
    __shared__ float sWt[SWT_N];
    __shared__ float sStage[WAVES_PER_BLOCK * WAVE_LDS];

    const int tid  = threadIdx.x;
    const int wave = tid >> 5;
    const int lane = tid & 31;
    const int m    = lane & 15;   // M (A rows) / N (B,C cols)
    const int kg   = lane >> 4;   // k-subgroup

    // ---- stage zero-padded, K-permuted weights once per block ----
    for (int i = tid; i < SW1_N; i += 256) {
        const int r = i >> 5, c = i & 31;
        float v = 0.0f;
        if (r < 16)       v = W1[(r + 1) * 32 + c];
        else if (r == 16) v = W1[c];
        sWt[SW1_OFF + i] = v;
    }
    for (int i = tid; i < SW2_N; i += 256) sWt[SW2_OFF + i] = W2[i];
    if (tid < 32) sWt[SB1_OFF + tid] = b1[tid];
    if (tid < 16) sWt[SB2_OFF + tid] = b2[tid];
    if (tid < 16) sWt[SW3_OFF + tid] = W3[tid];
    __syncthreads();

    // ---- hoist B fragments + biases into registers (reused across 8 tiles) ----
    v2f B1a[5], B1b[5], B2f[8];
    #pragma unroll
    for (int t = 0; t < 5; ++t) {
        const int k0 = 4 * t + 2 * kg;
        B1a[t].x = sWt[SW1_OFF + (k0    ) * 32 + m];
        B1a[t].y = sWt[SW1_OFF + (k0 + 1) * 32 + m];
        B1b[t].x = sWt[SW1_OFF + (k0    ) * 32 + 16 + m];
        B1b[t].y = sWt[SW1_OFF + (k0 + 1) * 32 + 16 + m];
    }
    #pragma unroll
    for (int t = 0; t < 8; ++t) {
        const int k0 = 4 * t + 2 * kg;
        B2f[t].x = sWt[SW2_OFF + (k0    ) * 16 + m];
        B2f[t].y = sWt[SW2_OFF + (k0 + 1) * 16 + m];
    }
    const float biasA = sWt[SB1_OFF + m];
    const float biasB = sWt[SB1_OFF + 16 + m];
    const float bias2 = sWt[SB2_OFF + m];
    const float b3v   = b3[0];
    const float qp1 = q_params[1], qp2 = q_params[2], qp3 = q_params[3],
                qp4 = q_params[4], qp5 = q_params[5], qp6 = q_params[6];

    float* wA  = &sStage[wave * WAVE_LDS];
    float* wH1 = wA + A_FLOATS;
    float* wH2 = wH1 + H1_FLOATS;

    // A-tile K padding (cols 17..19) written once; never overwritten
    if (lane < 16) {
        wA[m * STRIDE_A + 17] = 0.0f;
        wA[m * STRIDE_A + 18] = 0.0f;
        wA[m * STRIDE_A + 19] = 0.0f;
    }

    const int tile0 = (blockIdx.x * WAVES_PER_BLOCK + wave) * TILES_PER_WAVE;
    const v4f* xc4 = (const v4f*)x_c;

    for (int t = 0; t < TILES_PER_WAVE; ++t) {
        const int row0 = (tile0 + t) * ROWS_PER_TILE;

        // ---- phase 0: activations. x_c tile = 1024B contiguous -> 2x b128/lane
        #pragma unroll
        for (int j = 0; j < 2; ++j) {
            const int i4 = lane + 32 * j;            // 0..63 float4s
            const v4f v = xc4[row0 * 4 + i4];
            const int rr = i4 >> 2, cc = (i4 & 3) * 4;
            *(v4f*)&wA[rr * STRIDE_A + cc] = v;      // 16B-aligned (stride 20)
        }
        // quantum part collapses analytically: <Z0> = prod cos(x_q[:,i]+qp[i]), i=1..6
        if (lane < 16) {
            const float* xr = x_q + (size_t)(row0 + m) * 7;
            const float q = __cosf(xr[1] + qp1) * __cosf(xr[2] + qp2)
                          * __cosf(xr[3] + qp3) * __cosf(xr[4] + qp4)
                          * __cosf(xr[5] + qp5) * __cosf(xr[6] + qp6);
            wA[m * STRIDE_A + 16] = q;
        }
        __syncthreads();

        // ---- layer 1: [16x20] @ W1p[20x32], bias pre-loaded into C ----
        v8f c1a, c1b;
        #pragma unroll
        for (int v = 0; v < 8; ++v) { c1a[v] = biasA; c1b[v] = biasB; }
        #pragma unroll
        for (int kt = 0; kt < 5; ++kt) {
            const int k0 = 4 * kt + 2 * kg;
            v2f a;
            a.x = wA[m * STRIDE_A + k0];
            a.y = wA[m * STRIDE_A + k0 + 1];
            c1a = __builtin_amdgcn_wmma_f32_16x16x4_f32(false, a, false, B1a[kt],
                                                        (short)0, c1a, false, false);
            c1b = __builtin_amdgcn_wmma_f32_16x16x4_f32(false, a, false, B1b[kt],
                                                        (short)0, c1b, false, false);
        }
        // ReLU fused into the C->A layout staging store
        #pragma unroll
        for (int v = 0; v < 8; ++v) {
            wH1[(v + 8 * kg) * STRIDE_H1 + m]      = fmaxf(c1a[v], 0.0f);
            wH1[(v + 8 * kg) * STRIDE_H1 + 16 + m] = fmaxf(c1b[v], 0.0f);
        }
        __syncthreads();

        // ---- layer 2: [16x32] @ W2[32x16] ----
        v8f c2;
        #pragma unroll
        for (int v = 0; v < 8; ++v) c2[v] = bias2;
        #pragma unroll
        for (int kt = 0; kt < 8; ++kt) {
            const int k0 = 4 * kt + 2 * kg;
            v2f a;
            a.x = wH1[m * STRIDE_H1 + k0];
            a.y = wH1[m * STRIDE_H1 + k0 + 1];
            c2 = __builtin_amdgcn_wmma_f32_16x16x4_f32(false, a, false, B2f[kt],
                                                       (short)0, c2, false, false);
        }
        #pragma unroll
        for (int v = 0; v < 8; ++v)
            wH2[(v + 8 * kg) * STRIDE_H2 + m] = fmaxf(c2[v], 0.0f);
        __syncthreads();

        // ---- layer 3: [16x16] @ W3[16x1] + b3 (rows contiguous -> b128 loads) ----
        if (lane < 16) {
            float acc = b3v;
            #pragma unroll
            for (int n = 0; n < 16; ++n)
                acc = fmaf(wH2[m * STRIDE_H2 + n], sWt[SW3_OFF + n], acc);
            out[row0 + m] = acc;
        }
        // no trailing barrier needed: buffers are per-wave, DS ops are in-order
        // per wave, and two __syncthreads separate reuse across iterations
    }
}

extern "C" void kernel_launch(void* const* d_in, const int* in_sizes, int n_in,
                              void* d_out, int out_size, void* d_ws, size_t ws_size,
                              hipStream_t stream) {
    const float* x_q = (const float*)d_in[0];
    const float* x_c = (const float*)d_in[1];
    const float* qp  = (const float*)d_in[2];
    const float* W1  = (const float*)d_in[3];
    const float* b1  = (const float*)d_in[4];
    const float* W2  = (const float*)d_in[5];
    const float* b2  = (const float*)d_in[6];
    const float* W3  = (const float*)d_in[7];
    const float* b3  = (const float*)d_in[8];
    float* out = (float*)d_out;

    const int B = in_sizes[0] / 7;  // 262144
    const int rows_per_block = WAVES_PER_BLOCK * TILES_PER_WAVE * ROWS_PER_TILE; // 1024
    const int nblocks = B / rows_per_block;  // 256 (exact)

    hipLaunchKernelGGL(hybridq_kernel, dim3(nblocks), dim3(256), 0, stream,
                       x_q, x_c, qp, W1, b1, W2, b2, W3, b3, out);
}